// SACIndustrialModel_50534585204789
// MI455X (gfx1250) — compile-verified
//
#include <hip/hip_runtime.h>

#define D   256
#define Bsz 64
#define Tsz 128
#define Wsz 32
#define OUT1 (Bsz * Wsz * Tsz)   // 262144, size of one output tensor

typedef float v2f __attribute__((ext_vector_type(2)));
typedef float v8f __attribute__((ext_vector_type(8)));

// ---------------------------------------------------------------------------
// One GEMM stage over a 16-row tile using V_WMMA_F32_16X16X4_F32 (exact f32).
//   src : LDS tile [16][258] (padded: conflict-free, 8B-aligned for k%4==0)
//   Wg  : global weights [K][256] row-major
//   dst : LDS tile [16][258] (or null), gout: global [16][256] (or null)
// Wave wv (of 16) owns output columns n0..n0+15.
// A-frag (16x4 f32): lanes 0-15 -> M=lane, v0=K+0,v1=K+1; lanes 16-31 -> K+2,K+3
// B-frag (4x16 f32): lanes 0-15 -> N=lane, v0=K+0,v1=K+1; lanes 16-31 -> K+2,K+3
// C/D   : VGPR r -> M=r (lanes 0-15) / M=r+8 (lanes 16-31), N = n0 + (lane&15)
// ---------------------------------------------------------------------------
__device__ __forceinline__ void gemm_stage(
    const float* src, int K, const float* __restrict__ Wg,
    const float* __restrict__ bias, bool relu,
    float* dst, float* __restrict__ gout)
{
  const int tid  = threadIdx.x;
  const int lane = tid & 31;
  const int wv   = tid >> 5;        // 0..15
  const int nl   = lane & 15;
  const int hi   = lane >> 4;       // 0 or 1
  const int n0   = wv * 16;

  v8f acc = {};
  for (int kk = 0; kk < K; kk += 4) {
    const int ka = kk + hi * 2;
    v2f a = *(const v2f*)(src + nl * 258 + ka);      // ds_load_b64
    v2f b;
    b.x = Wg[(size_t)ka * D + n0 + nl];              // coalesced 64B / half-wave
    b.y = Wg[(size_t)(ka + 1) * D + n0 + nl];
    acc = __builtin_amdgcn_wmma_f32_16x16x4_f32(
        false, a, false, b, (short)0, acc, false, false);
  }
  #pragma unroll
  for (int r = 0; r < 8; ++r) {
    const int row = r + hi * 8;
    const int col = n0 + nl;
    float v = acc[r];
    if (bias) v += bias[col];
    if (relu) v = fmaxf(v, 0.0f);
    if (dst)  dst[row * 258 + col] = v;
    if (gout) gout[(size_t)row * D + col] = v;
  }
}

// ---------------------------------------------------------------------------
// Full per-row chain:  X[16,KIN] -> relu(X@W1+b1) -> @W2+b2 (=F)
//   branch 1: (F@aW+ab)@fW        -> proj_out
//   branch 2:  F@qW1              -> q1_out        (biases folded in later)
//   branch 3:  F@qW2              -> q2_out
// ---------------------------------------------------------------------------
template<int KIN>
__global__ __launch_bounds__(512)
void chain_kernel(const float* __restrict__ X,
                  const float* __restrict__ W1, const float* __restrict__ b1,
                  const float* __restrict__ W2, const float* __restrict__ b2,
                  const float* __restrict__ aW, const float* __restrict__ ab,
                  const float* __restrict__ fW,
                  float* __restrict__ proj_out,
                  const float* __restrict__ qW1, float* __restrict__ q1_out,
                  const float* __restrict__ qW2, float* __restrict__ q2_out)
{
  __shared__ float bufA[16 * 258];
  __shared__ float bufB[16 * 258];
  const int    tid = threadIdx.x;
  const size_t r0  = (size_t)blockIdx.x * 16;

  for (int i = tid; i < 16 * KIN; i += 512) {
    const int r = i / KIN, c = i % KIN;
    bufA[r * 258 + c] = X[(r0 + r) * KIN + c];
  }
  __syncthreads();
  gemm_stage(bufA, KIN, W1, b1, true,  bufB, nullptr);           // hidden
  __syncthreads();
  gemm_stage(bufB, D,   W2, b2, false, bufA, nullptr);           // F -> bufA
  __syncthreads();
  gemm_stage(bufA, D,   aW, ab, false, bufB, nullptr);           // pre -> bufB
  __syncthreads();
  gemm_stage(bufB, D,   fW,  nullptr, false, nullptr, proj_out + r0 * D);
  gemm_stage(bufA, D,   qW1, nullptr, false, nullptr, q1_out   + r0 * D);
  gemm_stage(bufA, D,   qW2, nullptr, false, nullptr, q2_out   + r0 * D);
}

// ---------------------------------------------------------------------------
// Fused heads: for each (b, w, t):
//   mean = mask * (sum_d relu(tproj+wproj+f1b)           * f2W + f2b)
//   qX   =         sum_d relu(qtX + qwX + act*waX + bX1) * WX2 + bX2
//   std  = mask * exp(log_std)
// Block: one b and one 32-wide t-tile. lane = t, wave handles 4 w rows.
// ---------------------------------------------------------------------------
__global__ __launch_bounds__(256)
void fuse_kernel(const float* __restrict__ tproj, const float* __restrict__ wproj,
                 const float* __restrict__ qt1,   const float* __restrict__ qw1,
                 const float* __restrict__ qt2,   const float* __restrict__ qw2,
                 const float* __restrict__ action, const float* __restrict__ valid_mask,
                 const float* __restrict__ log_std,
                 const float* __restrict__ a_f1_b, const float* __restrict__ a_f2_W,
                 const float* __restrict__ a_f2_b,
                 const float* __restrict__ q1_b1, const float* __restrict__ q1_W2,
                 const float* __restrict__ q1_b2, const float* __restrict__ q1_wa,
                 const float* __restrict__ q2_b1, const float* __restrict__ q2_W2,
                 const float* __restrict__ q2_b2, const float* __restrict__ q2_wa,
                 float* __restrict__ out)
{
  extern __shared__ float smem[];
  float* sT   = smem;              // [32][257]  task side, padded (banks: l*257 ≡ l)
  float* sW   = sT + 32 * 257;     // [32][256]  worker side (broadcast reads)
  float* sB   = sW + 32 * 256;     // [256] hidden bias
  float* sV   = sB + 256;          // [256] output weight column
  float* sA   = sV + 256;          // [256] action weight row (0 for mean head)
  float* sAct = sA + 256;          // [32][32] action tile

  const int tid  = threadIdx.x;    // 256
  const int lane = tid & 31;       // t within tile
  const int wv   = tid >> 5;       // 0..7
  const int b    = blockIdx.x >> 2;
  const int t0   = (blockIdx.x & 3) * 32;

  const float mval = valid_mask[b * Tsz + t0 + lane];

  // std = exp(log_std[w,t]) * mask  (log_std shape [1,W,T])
  #pragma unroll
  for (int j = 0; j < 4; ++j) {
    const int w = wv * 4 + j;
    out[(size_t)OUT1 + (size_t)b * Wsz * Tsz + w * Tsz + t0 + lane] =
        expf(log_std[w * Tsz + t0 + lane]) * mval;
  }

  for (int i = tid; i < 32 * 32; i += 256) {
    const int w = i >> 5, t = i & 31;
    sAct[i] = action[(size_t)b * Wsz * Tsz + w * Tsz + t0 + t];
  }

  const float* Tps[3]  = {tproj,  qt1,   qt2};
  const float* Wps[3]  = {wproj,  qw1,   qw2};
  const float* Bps[3]  = {a_f1_b, q1_b1, q2_b1};
  const float* Vps[3]  = {a_f2_W, q1_W2, q2_W2};
  const float* b2s[3]  = {a_f2_b, q1_b2, q2_b2};
  const float* Aps[3]  = {nullptr, q1_wa, q2_wa};
  const size_t offs[3] = {0, 2ull * OUT1, 3ull * OUT1};

  for (int n = 0; n < 3; ++n) {
    __syncthreads();
    for (int i = tid; i < 32 * 256; i += 256) {          // task tile, padded rows
      const int t = i >> 8, d = i & 255;
      sT[t * 257 + d] = Tps[n][((size_t)b * Tsz + t0 + t) * D + d];
    }
    for (int i = tid; i < 32 * 256; i += 256)            // worker tile (contiguous)
      sW[i] = Wps[n][(size_t)b * Wsz * D + i];
    sB[tid] = Bps[n][tid];
    sV[tid] = Vps[n][tid];
    sA[tid] = Aps[n] ? Aps[n][tid] : 0.0f;
    __syncthreads();

    const float c2 = b2s[n][0];
    #pragma unroll
    for (int j = 0; j < 4; ++j) {
      const int   w  = wv * 4 + j;
      const float av = sAct[w * 32 + lane];
      float acc = 0.0f;
      #pragma unroll 4
      for (int d = 0; d < 256; ++d) {
        float h = sT[lane * 257 + d] + sW[w * 256 + d] + sB[d];
        h = fmaf(av, sA[d], h);
        h = fmaxf(h, 0.0f);
        acc = fmaf(h, sV[d], acc);
      }
      float o = acc + c2;
      if (n == 0) o *= mval;
      out[offs[n] + (size_t)b * Wsz * Tsz + w * Tsz + t0 + lane] = o;
    }
  }
}

// ---------------------------------------------------------------------------
extern "C" void kernel_launch(void* const* d_in, const int* in_sizes, int n_in,
                              void* d_out, int out_size, void* d_ws, size_t ws_size,
                              hipStream_t stream) {
  (void)in_sizes; (void)n_in; (void)out_size; (void)ws_size;
  const float* task_obs   = (const float*)d_in[0];
  const float* worker_obs = (const float*)d_in[1];
  const float* action     = (const float*)d_in[2];
  const float* valid_mask = (const float*)d_in[3];
  const float* te_W1 = (const float*)d_in[4];
  const float* te_b1 = (const float*)d_in[5];
  const float* te_W2 = (const float*)d_in[6];
  const float* te_b2 = (const float*)d_in[7];
  const float* we_W1 = (const float*)d_in[8];
  const float* we_b1 = (const float*)d_in[9];
  const float* we_W2 = (const float*)d_in[10];
  const float* we_b2 = (const float*)d_in[11];
  const float* a_t_W = (const float*)d_in[12];
  const float* a_t_b = (const float*)d_in[13];
  const float* a_w_W = (const float*)d_in[14];
  const float* a_w_b = (const float*)d_in[15];
  const float* a_f1_W = (const float*)d_in[16];
  const float* a_f1_b = (const float*)d_in[17];
  const float* a_f2_W = (const float*)d_in[18];
  const float* a_f2_b = (const float*)d_in[19];
  const float* log_std = (const float*)d_in[20];
  const float* q1_W1 = (const float*)d_in[21];
  const float* q1_b1 = (const float*)d_in[22];
  const float* q1_W2 = (const float*)d_in[23];
  const float* q1_b2 = (const float*)d_in[24];
  const float* q2_W1 = (const float*)d_in[25];
  const float* q2_b1 = (const float*)d_in[26];
  const float* q2_W2 = (const float*)d_in[27];
  const float* q2_b2 = (const float*)d_in[28];

  // workspace: task-side [8192,256] x3, worker-side [2048,256] x3  (~30 MB)
  float* tproj = (float*)d_ws;
  float* qt1   = tproj + (size_t)8192 * 256;
  float* qt2   = qt1   + (size_t)8192 * 256;
  float* wproj = qt2   + (size_t)8192 * 256;
  float* qw1   = wproj + (size_t)2048 * 256;
  float* qw2   = qw1   + (size_t)2048 * 256;

  // task rows: 8192 / 16 = 512 tiles
  chain_kernel<64><<<512, 512, 0, stream>>>(
      task_obs, te_W1, te_b1, te_W2, te_b2, a_t_W, a_t_b,
      a_f1_W /* f1t = rows 0..255 */, tproj,
      q1_W1 /* wt */, qt1, q2_W1, qt2);

  // worker rows: 2048 / 16 = 128 tiles
  chain_kernel<32><<<128, 512, 0, stream>>>(
      worker_obs, we_W1, we_b1, we_W2, we_b2, a_w_W, a_w_b,
      a_f1_W + (size_t)256 * 256 /* f1w */, wproj,
      q1_W1 + (size_t)256 * 256 /* ww */, qw1,
      q2_W1 + (size_t)256 * 256, qw2);

  const size_t lds = (size_t)(32 * 257 + 32 * 256 + 3 * 256 + 32 * 32) * sizeof(float);
  fuse_kernel<<<Bsz * (Tsz / 32), 256, lds, stream>>>(
      tproj, wproj, qt1, qw1, qt2, qw2,
      action, valid_mask, log_std,
      a_f1_b, a_f2_W, a_f2_b,
      q1_b1, q1_W2, q1_b2, q1_W1 + (size_t)512 * 256 /* wa */,
      q2_b1, q2_W2, q2_b2, q2_W1 + (size_t)512 * 256,
      (float*)d_out);
}